// NGCF_2894807958110
// MI455X (gfx1250) — compile-verified
//
#include <hip/hip_runtime.h>
#include <cstdint>

#define N_USER  100000
#define N_ITEM  100000
#define N_NODES 200000
#define EMB     64
#define BATCH   1024
#define NNZ_CNT 6400000

typedef __attribute__((ext_vector_type(2))) float v2f;
typedef __attribute__((ext_vector_type(8))) float v8f;

// ---------------------------------------------------------------------------
// SpMM: y[row,:] += val * x[col,:]   (rows sorted ascending; y pre-zeroed)
// One wave per SPMM_CHUNK nnz. Lane owns columns {lane, lane+32}.
// Register accumulation while row unchanged; atomic flush on row change.
// ---------------------------------------------------------------------------
#define SPMM_CHUNK 256
__global__ __launch_bounds__(256) void spmm_kernel(
    const int* __restrict__ row, const int* __restrict__ col,
    const float* __restrict__ val, const float* __restrict__ x,
    float* __restrict__ y, int nnz)
{
  const int  lane = threadIdx.x & 31;
  const long wave = ((long)blockIdx.x * blockDim.x + threadIdx.x) >> 5;
  long start = wave * SPMM_CHUNK;
  if (start >= nnz) return;
  long end = start + SPMM_CHUNK;
  if (end > nnz) end = nnz;

  float acc0 = 0.f, acc1 = 0.f;
  int cur = row[start];                       // uniform scalar load
  for (long base = start; base < end; base += 32) {
    long e = base + lane;
    int r = -1, c = 0; float v = 0.f;
    if (e < end) { r = row[e]; c = col[e]; v = val[e]; }
#pragma unroll 1
    for (int i = 0; i < 32; ++i) {
      int ri = __shfl(r, i);
      if (ri < 0) break;                      // past end of chunk
      int   ci = __shfl(c, i);
      float vi = __shfl(v, i);
      if (ri != cur) {
        atomicAdd(&y[(long)cur * EMB + lane],      acc0);
        atomicAdd(&y[(long)cur * EMB + lane + 32], acc1);
        acc0 = 0.f; acc1 = 0.f; cur = ri;
      }
      const float* xp = x + (long)ci * EMB;   // L2-resident (ego = 51MB < 192MB)
      acc0 += vi * xp[lane];
      acc1 += vi * xp[lane + 32];
    }
  }
  atomicAdd(&y[(long)cur * EMB + lane],      acc0);
  atomicAdd(&y[(long)cur * EMB + lane + 32], acc1);
}

// ---------------------------------------------------------------------------
// Fused NGCF layer: per wave, 16 rows x 64 cols of
//   d = leaky_relu(side@Wgc + bgc + (ego*side)@Wbi + bbi)
//   ego_next = d ; normed = d / max(||d||_row, eps)
// V_WMMA_F32_16X16X4_F32, K = 64 in 16 steps, 4 N-tiles, 2 GEMMs.
// Weights staged in LDS K-pair interleaved: lds[k/2][n] = {W[k][n], W[k+1][n]}
// so each lane's B fragment is one aligned ds_load_b64 (no repacking movs).
// A layout (16x4 f32): lanes 0-15 -> K={0,1}, lanes 16-31 -> K={2,3}.
// C/D layout: VGPR j -> row j (lanes 0-15) / row j+8 (lanes 16-31), N = lane&15.
// ---------------------------------------------------------------------------
__global__ __launch_bounds__(256) void ngcf_layer_kernel(
    const float* __restrict__ side, const float* __restrict__ ego,
    const float* __restrict__ Wgc,  const float* __restrict__ bgc,
    const float* __restrict__ Wbi,  const float* __restrict__ bbi,
    float* __restrict__ ego_next,   float* __restrict__ normed, int M)
{
  __shared__ v2f   lds_wgc2[(EMB / 2) * EMB];   // 16 KB, [k/2][n] pairs
  __shared__ v2f   lds_wbi2[(EMB / 2) * EMB];   // 16 KB
  __shared__ float lds_bias[EMB];               // combined bias

  // stage interleaved: idx -> (k2 = idx>>6, n = idx&63); coalesced across n
  for (int idx = threadIdx.x; idx < (EMB / 2) * EMB; idx += 256) {
    const int k2 = idx >> 6;
    const int n  = idx & 63;
    v2f g, b;
    g.x = Wgc[(2 * k2)     * EMB + n];
    g.y = Wgc[(2 * k2 + 1) * EMB + n];
    b.x = Wbi[(2 * k2)     * EMB + n];
    b.y = Wbi[(2 * k2 + 1) * EMB + n];
    lds_wgc2[idx] = g;
    lds_wbi2[idx] = b;
  }
  if (threadIdx.x < EMB)
    lds_bias[threadIdx.x] = bgc[threadIdx.x] + bbi[threadIdx.x];
  __syncthreads();

  const int lane  = threadIdx.x & 31;
  const int gwave = (int)((blockIdx.x * blockDim.x + threadIdx.x) >> 5);
  const int row0  = gwave * 16;
  if (row0 >= M) return;                      // after barrier; whole-wave uniform

  const int mrow = row0 + (lane & 15);
  const int ksel = (lane >> 4) << 1;          // 0 or 2
  const int ncol = lane & 15;

  v8f c1[4] = {};                             // side @ Wgc tiles (N = t*16..)
  v8f c2[4] = {};                             // (ego*side) @ Wbi tiles

#pragma unroll
  for (int kk = 0; kk < 16; ++kk) {
    const int kbase = kk * 4 + ksel;          // even
    const int k2row = (kbase >> 1) * EMB;     // interleaved LDS row base
    v2f s  = *(const v2f*)(side + (long)mrow * EMB + kbase);
    v2f eg = *(const v2f*)(ego  + (long)mrow * EMB + kbase);
    v2f p; p.x = s.x * eg.x; p.y = s.y * eg.y;
#pragma unroll
    for (int t = 0; t < 4; ++t) {
      const int n = t * 16 + ncol;
      v2f bg = lds_wgc2[k2row + n];           // single ds_load_b64
      v2f bb = lds_wbi2[k2row + n];
      c1[t] = __builtin_amdgcn_wmma_f32_16x16x4_f32(false, s, false, bg,
                                                    (short)0, c1[t], false, false);
      c2[t] = __builtin_amdgcn_wmma_f32_16x16x4_f32(false, p, false, bb,
                                                    (short)0, c2[t], false, false);
    }
  }

  // bias + leaky relu
  float d[4][8];
#pragma unroll
  for (int t = 0; t < 4; ++t) {
    const int   n    = t * 16 + ncol;
    const float bsum = lds_bias[n];
#pragma unroll
    for (int j = 0; j < 8; ++j) {
      float v = c1[t][j] + c2[t][j] + bsum;
      d[t][j] = v > 0.f ? v : 0.2f * v;
    }
  }

  // row L2 norm + stores. Row of slot j is row0 + (lane<16 ? j : j+8);
  // xor-reduce over masks 1,2,4,8 stays within each 16-lane half.
#pragma unroll
  for (int j = 0; j < 8; ++j) {
    float ss = 0.f;
#pragma unroll
    for (int t = 0; t < 4; ++t) ss += d[t][j] * d[t][j];
    ss += __shfl_xor(ss, 1);
    ss += __shfl_xor(ss, 2);
    ss += __shfl_xor(ss, 4);
    ss += __shfl_xor(ss, 8);
    const float inv = 1.f / fmaxf(sqrtf(ss), 1e-12f);
    const int   r   = row0 + ((lane < 16) ? j : j + 8);
#pragma unroll
    for (int t = 0; t < 4; ++t) {
      const int n = t * 16 + ncol;
      ego_next[(long)r * EMB + n] = d[t][j];
      normed  [(long)r * EMB + n] = d[t][j] * inv;
    }
  }
}

// ---------------------------------------------------------------------------
// Scatter 64-column layer slice of the selected 3*1024 rows into d_out(3072x256)
// ---------------------------------------------------------------------------
__global__ __launch_bounds__(64) void gather_kernel(
    const float* __restrict__ src, const int* __restrict__ users,
    const int* __restrict__ pos,   const int* __restrict__ neg,
    float* __restrict__ out, int col_off)
{
  const int j = blockIdx.x;                   // 0..3071
  const int c = threadIdx.x;                  // 0..63
  int node;
  if (j < BATCH)          node = users[j];
  else if (j < 2 * BATCH) node = N_USER + pos[j - BATCH];
  else                    node = N_USER + neg[j - 2 * BATCH];
  out[(size_t)j * (4 * EMB) + col_off + c] = src[(size_t)node * EMB + c];
}

// ---------------------------------------------------------------------------
extern "C" void kernel_launch(void* const* d_in, const int* in_sizes, int n_in,
                              void* d_out, int out_size, void* d_ws, size_t ws_size,
                              hipStream_t stream)
{
  (void)in_sizes; (void)n_in; (void)out_size; (void)ws_size;

  const int*   users = (const int*)d_in[0];
  const int*   pos   = (const int*)d_in[1];
  const int*   neg   = (const int*)d_in[2];
  const int*   arow  = (const int*)d_in[3];
  const int*   acol  = (const int*)d_in[4];
  const float* aval  = (const float*)d_in[5];
  const float* uemb  = (const float*)d_in[6];
  const float* iemb  = (const float*)d_in[7];
  const float* Wgc[3] = {(const float*)d_in[8],  (const float*)d_in[12], (const float*)d_in[16]};
  const float* bgc[3] = {(const float*)d_in[9],  (const float*)d_in[13], (const float*)d_in[17]};
  const float* Wbi[3] = {(const float*)d_in[10], (const float*)d_in[14], (const float*)d_in[18]};
  const float* bbi[3] = {(const float*)d_in[11], (const float*)d_in[15], (const float*)d_in[19]};
  float* out = (float*)d_out;

  const size_t NODE_ELEMS = (size_t)N_NODES * EMB;      // 12.8M floats
  float* egoA = (float*)d_ws;
  float* side = egoA + NODE_ELEMS;
  float* egoB = side + NODE_ELEMS;
  float* nrm  = egoB + NODE_ELEMS;                      // total 204.8 MB

  // ego = concat(user_emb, item_emb)
  hipMemcpyAsync(egoA, uemb, (size_t)N_USER * EMB * sizeof(float),
                 hipMemcpyDeviceToDevice, stream);
  hipMemcpyAsync(egoA + (size_t)N_USER * EMB, iemb,
                 (size_t)N_ITEM * EMB * sizeof(float),
                 hipMemcpyDeviceToDevice, stream);

  // layer-0 slice of output = initial ego rows
  gather_kernel<<<3 * BATCH, EMB, 0, stream>>>(egoA, users, pos, neg, out, 0);

  const int spmm_waves  = (NNZ_CNT + SPMM_CHUNK - 1) / SPMM_CHUNK;
  const int spmm_blocks = (spmm_waves * 32 + 255) / 256;
  const int gemm_waves  = (N_NODES + 15) / 16;
  const int gemm_blocks = (gemm_waves * 32 + 255) / 256;

  float* cur = egoA;
  float* nxt = egoB;
  for (int k = 0; k < 3; ++k) {
    hipMemsetAsync(side, 0, NODE_ELEMS * sizeof(float), stream);
    spmm_kernel<<<spmm_blocks, 256, 0, stream>>>(arow, acol, aval, cur, side, NNZ_CNT);
    ngcf_layer_kernel<<<gemm_blocks, 256, 0, stream>>>(
        side, cur, Wgc[k], bgc[k], Wbi[k], bbi[k], nxt, nrm, N_NODES);
    gather_kernel<<<3 * BATCH, EMB, 0, stream>>>(nrm, users, pos, neg, out,
                                                 (k + 1) * EMB);
    float* tmp = cur; cur = nxt; nxt = tmp;
  }
}